// GraphConvLayer_21552145891947
// MI455X (gfx1250) — compile-verified
//
#include <hip/hip_runtime.h>
#include <hip/hip_bf16.h>
#include <math.h>

#define B_   8
#define S_   128
#define D_   300
#define E_   50
#define DEP_ 50
#define EPS_ 1e-6f
#define NT_  19   // ceil(D/16)

typedef __attribute__((ext_vector_type(2))) float v2f;
typedef __attribute__((ext_vector_type(8))) float v8f;

// ---------------------------------------------------------------------------
// K1: Abar[b,s,t] = (1/E) * ( sum_e wps[b,s,t,e] + sum_e sl[b,e,s,t] )
// grid = B*S blocks, 128 threads (t). sl reads coalesced across t; wps reads
// 50 contiguous floats per lane via float2.
// ---------------------------------------------------------------------------
__global__ void __launch_bounds__(S_) k_abar(const float* __restrict__ wps,
                                             const float* __restrict__ sl,
                                             float* __restrict__ abar) {
    const int bs = blockIdx.x;           // b*S + s
    const int t  = threadIdx.x;          // 0..127
    const int b  = bs >> 7;
    const int s  = bs & (S_ - 1);

    float acc = 0.f;
    const float2* w2 = reinterpret_cast<const float2*>(wps + ((size_t)bs * S_ + t) * E_);
#pragma unroll
    for (int e = 0; e < E_ / 2; ++e) { float2 v = w2[e]; acc += v.x + v.y; }

    const float* slp = sl + ((size_t)b * E_ * S_ + s) * S_ + t;   // stride S*S over e
#pragma unroll 5
    for (int e = 0; e < E_; ++e) acc += slp[(size_t)e * S_ * S_];

    abar[(size_t)bs * S_ + t] = acc * (1.0f / E_);
}

// ---------------------------------------------------------------------------
// K2: Ax[b] = Abar[b] (128x128) @ X[b] (128x300), f32 WMMA 16x16x4.
// One wave per 16x16 output tile; grid = B * 8 * 19, block = 32.
// Padded N-columns are handled by CLAMPING the load index (no exec-mask
// divergence in the inner loop); garbage columns are discarded at store.
// A frag (16x4): lane m = l&15, kpair = (l>>4)*2.  B frag (4x16): lane n.
// C/D: lane n = l&15, VGPR r -> row r + (l>>4)*8.
// ---------------------------------------------------------------------------
__global__ void __launch_bounds__(32) k_gemm1(const float* __restrict__ abar,
                                              const float* __restrict__ x,
                                              float* __restrict__ ax) {
    int id = blockIdx.x;
    const int nt = id % NT_;  id /= NT_;
    const int mt = id % (S_ / 16);
    const int b  = id / (S_ / 16);
    const int lane = threadIdx.x;
    const int hl   = lane >> 4;        // which K-pair half
    const int mn   = lane & 15;        // M for A-frag, N for B/C-frag
    const int n    = nt * 16 + mn;
    const int nc   = (n < D_) ? n : (D_ - 1);   // clamped: always a valid address

    const float* Arow = abar + (size_t)b * S_ * S_ + (size_t)(mt * 16 + mn) * S_;
    const float* Bcol = x + (size_t)b * S_ * D_ + nc;

    v8f c = {};
#pragma unroll 4
    for (int k = 0; k < S_; k += 4) {
        const int ka = k + hl * 2;
        const float2 ap = *reinterpret_cast<const float2*>(Arow + ka);   // contiguous pair
        v2f a;  a.x = ap.x;  a.y = ap.y;
        v2f bv; bv.x = Bcol[(size_t)ka * D_];  bv.y = Bcol[(size_t)(ka + 1) * D_];
        c = __builtin_amdgcn_wmma_f32_16x16x4_f32(false, a, false, bv, (short)0, c, false, false);
    }
    if (n < D_) {
#pragma unroll
        for (int r = 0; r < 8; ++r) {
            const int mo = mt * 16 + hl * 8 + r;
            ax[((size_t)b * S_ + mo) * D_ + n] = c[r];
        }
    }
}

// ---------------------------------------------------------------------------
// K3: g = Ax @ W_w^T + W_b ; LayerNorm(ddof=1, ln_a,ln_b, std+eps) ; ReLU.
// Block = 128 threads (4 waves) owns a 16-row strip; waves split 19 N-tiles.
// Tile accumulated with f32 WMMA (K=300 -> 75 steps); B-fragment pair is
// contiguous in W_w (row n, cols ka,ka+1) -> single float2 load. Clamped
// index for padded columns keeps the inner loop divergence-free.
// ---------------------------------------------------------------------------
__global__ void __launch_bounds__(128) k_gemm2_ln(const float* __restrict__ ax,
                                                  const float* __restrict__ ww,
                                                  const float* __restrict__ wb,
                                                  const float* __restrict__ lna,
                                                  const float* __restrict__ lnb,
                                                  float* __restrict__ node) {
    const int mt   = blockIdx.x % (S_ / 16);
    const int b    = blockIdx.x / (S_ / 16);
    const int tid  = threadIdx.x;
    const int wave = tid >> 5;
    const int lane = tid & 31;
    const int hl   = lane >> 4;
    const int mn   = lane & 15;

    __shared__ float gt[16][304];
    __shared__ float rmean[16], rinv[16];

    const float* Arow = ax + ((size_t)b * S_ + mt * 16 + mn) * D_;

    for (int nt = wave; nt < NT_; nt += 4) {
        const int n  = nt * 16 + mn;
        const int nc = (n < D_) ? n : (D_ - 1);
        const float* Brow = ww + (size_t)nc * D_;      // B[k][n] = W_w[n][k]
        v8f c = {};
#pragma unroll 4
        for (int k = 0; k < D_; k += 4) {              // D=300, exactly 75 steps
            const int ka = k + hl * 2;
            const float2 ap = *reinterpret_cast<const float2*>(Arow + ka);
            const float2 bp = *reinterpret_cast<const float2*>(Brow + ka);
            v2f a;  a.x = ap.x;  a.y = ap.y;
            v2f bv; bv.x = bp.x; bv.y = bp.y;
            c = __builtin_amdgcn_wmma_f32_16x16x4_f32(false, a, false, bv, (short)0, c, false, false);
        }
        if (n < D_) {
            const float bias = wb[n];
#pragma unroll
            for (int r = 0; r < 8; ++r) gt[hl * 8 + r][n] = c[r] + bias;
        }
    }
    __syncthreads();

    if (tid < 16) {
        float s = 0.f;
        for (int j = 0; j < D_; ++j) s += gt[tid][j];
        const float m = s / (float)D_;
        float v = 0.f;
        for (int j = 0; j < D_; ++j) { const float d = gt[tid][j] - m; v += d * d; }
        rmean[tid] = m;
        rinv[tid]  = 1.f / (sqrtf(v / (float)(D_ - 1)) + EPS_);   // std ddof=1, (std+eps)
    }
    __syncthreads();

    for (int idx = tid; idx < 16 * D_; idx += 128) {
        const int r = idx / D_, j = idx - r * D_;
        const float val = lna[j] * (gt[r][j] - rmean[r]) * rinv[r] + lnb[j];
        node[((size_t)b * S_ + mt * 16 + r) * D_ + j] = fmaxf(val, 0.f);
    }
}

// ---------------------------------------------------------------------------
// K4: pa[b,s,p] = node[b,s,:] . W1_w[p, :D] ; pb[b,s,p] = node[b,s,:] . W1_w[p, D:]
// Block per (b,s); node row cached in LDS; threads 0..49 -> pa, 64..113 -> pb.
// ---------------------------------------------------------------------------
__global__ void __launch_bounds__(128) k_pab(const float* __restrict__ node,
                                             const float* __restrict__ w1,
                                             float* __restrict__ pa,
                                             float* __restrict__ pb) {
    const int bs  = blockIdx.x;          // b*S + s
    const int tid = threadIdx.x;
    __shared__ float row[D_];
    for (int j = tid; j < D_; j += 128) row[j] = node[(size_t)bs * D_ + j];
    __syncthreads();

    int p = -1; const float* wrow = nullptr; float* out = nullptr;
    if (tid < DEP_)                      { p = tid;      wrow = w1 + (size_t)p * (2 * D_);       out = pa; }
    else if (tid >= 64 && tid < 64+DEP_) { p = tid - 64; wrow = w1 + (size_t)p * (2 * D_) + D_;  out = pb; }
    if (p >= 0) {
        float acc = 0.f;
#pragma unroll 4
        for (int j = 0; j < D_; ++j) acc += row[j] * wrow[j];
        out[(size_t)bs * DEP_ + p] = acc;
    }
}

// ---------------------------------------------------------------------------
// K5: edge[b,i,j,p] = pa[b,j,p] + pb[b,i,p] + W1_b[p]   (26 MB coalesced write)
// Block per (b,i); pb row + bias cached in LDS.
// ---------------------------------------------------------------------------
__global__ void __launch_bounds__(256) k_edge(const float* __restrict__ pa,
                                              const float* __restrict__ pb,
                                              const float* __restrict__ w1b,
                                              float* __restrict__ edge) {
    const int bi  = blockIdx.x;          // b*S + i
    const int b   = bi >> 7;
    const int tid = threadIdx.x;
    __shared__ float pbb[DEP_];
    if (tid < DEP_) pbb[tid] = pb[(size_t)bi * DEP_ + tid] + w1b[tid];
    __syncthreads();

    const float* pab  = pa + (size_t)b * S_ * DEP_;
    float*       erow = edge + (size_t)bi * S_ * DEP_;
    for (int idx = tid; idx < S_ * DEP_; idx += 256) {
        const int p = idx % DEP_;
        erow[idx] = pab[idx] + pbb[p];
    }
}

// ---------------------------------------------------------------------------
extern "C" void kernel_launch(void* const* d_in, const int* in_sizes, int n_in,
                              void* d_out, int out_size, void* d_ws, size_t ws_size,
                              hipStream_t stream) {
    (void)in_sizes; (void)n_in; (void)out_size; (void)ws_size;
    const float* wps = (const float*)d_in[0];   // (B,S,S,E)
    // d_in[1] = weight_adj : unused by the reference
    const float* gcn = (const float*)d_in[2];   // (B,S,D)
    const float* sl  = (const float*)d_in[3];   // (B,E,S,S)
    const float* ww  = (const float*)d_in[4];   // (D,D)
    const float* wb  = (const float*)d_in[5];   // (D,)
    const float* lna = (const float*)d_in[6];   // (D,)
    const float* lnb = (const float*)d_in[7];   // (D,)
    const float* w1  = (const float*)d_in[8];   // (DEP, 2D)
    const float* w1b = (const float*)d_in[9];   // (DEP,)

    float* node = (float*)d_out;                          // B*S*D
    float* edge = node + (size_t)B_ * S_ * D_;            // B*S*S*DEP

    float* ws   = (float*)d_ws;
    float* abar = ws;                                     // B*S*S      = 131072
    float* ax   = abar + (size_t)B_ * S_ * S_;            // B*S*D      = 307200
    float* pa   = ax   + (size_t)B_ * S_ * D_;            // B*S*DEP    =  51200
    float* pb   = pa   + (size_t)B_ * S_ * DEP_;          // B*S*DEP    =  51200

    k_abar    <<<B_ * S_,               S_,  0, stream>>>(wps, sl, abar);
    k_gemm1   <<<B_ * (S_/16) * NT_,    32,  0, stream>>>(abar, gcn, ax);
    k_gemm2_ln<<<B_ * (S_/16),          128, 0, stream>>>(ax, ww, wb, lna, lnb, node);
    k_pab     <<<B_ * S_,               128, 0, stream>>>(node, w1, pa, pb);
    k_edge    <<<B_ * S_,               256, 0, stream>>>(pa, pb, w1b, edge);
}